// LNODEFunc10_83159156785561
// MI455X (gfx1250) — compile-verified
//
#include <hip/hip_runtime.h>
#include <hip/hip_bf16.h>

#define F_M 20

// ---- CDNA5 feature guards (always compile-clean) --------------------------
#if defined(__gfx1250__) && __has_builtin(__builtin_amdgcn_global_load_async_to_lds_b128)
#define HAVE_ASYNC_LDS 1
#else
#define HAVE_ASYNC_LDS 0
#endif

#if defined(__gfx1250__) && __has_builtin(__builtin_amdgcn_wmma_f32_16x16x4_f32)
#define HAVE_WMMA_F32 1
#else
#define HAVE_WMMA_F32 0
#endif

#if HAVE_ASYNC_LDS
// Builtin prototype (from hipcc diagnostic): first param is AS1 int4*.
typedef int v4i_t __attribute__((vector_size(16)));
typedef __attribute__((address_space(1))) v4i_t* gptr_v4i;
typedef __attribute__((address_space(3))) v4i_t* lptr_v4i;
#endif

#if HAVE_WMMA_F32
typedef __attribute__((ext_vector_type(2))) float v2f;
typedef __attribute__((ext_vector_type(8))) float v8f;
#endif

// ---------------------------------------------------------------------------
// Kernel 1 (one wave, run once): fold the whole weight network into constants
//   out = cross(A*x, C*x) + beta*x + cu,  A = alpha*M1, C = M2.
// The four 20x20 @ 20 matvecs (c-vectors) are done with V_WMMA_F32_16X16X4_F32:
//   - A tile: W rows (M=0..15 / 16..19, zero-padded), K in 4-chunks (20 = 5*4)
//   - B tile: v in column N=0, zeros elsewhere
//   - accumulate over 5 K-tiles; read column 0 of D (lanes 0/16, vgpr 0..7)
// ---------------------------------------------------------------------------
__global__ void ln_ode_precompute(
    const float* __restrict__ t, const float* __restrict__ R, const float* __restrict__ m,
    const float* __restrict__ Wl1, const float* __restrict__ Wd1,
    const float* __restrict__ Wl2, const float* __restrict__ Wd2,
    const float* __restrict__ Wl3, const float* __restrict__ Wd3,
    const float* __restrict__ Wl4, const float* __restrict__ Wd4,
    const float* __restrict__ Wb1_lin, const float* __restrict__ Wb1_d1, const float* __restrict__ Wb1_d2,
    const float* __restrict__ Wb2_lin, const float* __restrict__ Wb2_d1, const float* __restrict__ Wb2_d2,
    const float* __restrict__ Wout, float* __restrict__ cst)
{
    __shared__ float cvec[4][32];   // c1, c2, c1u, c2u (only [0..19] used)
    const int lane = threadIdx.x;   // launched with exactly 32 threads, 1 block

#if HAVE_WMMA_F32
    {
        const float* Ws[4] = {Wb1_d1, Wb1_d2, Wb2_d1, Wb2_d2};
        const float* vs[4] = {Wb1_lin, Wb1_lin, Wb2_lin, Wb2_lin};
        const int n    = lane & 15;
        const int half = lane >> 4;           // 0: lanes 0-15, 1: lanes 16-31
        for (int mat = 0; mat < 4; ++mat) {
            const float* W = Ws[mat];
            const float* v = vs[mat];
            for (int mt = 0; mt < 2; ++mt) {
                v8f acc = {0.f,0.f,0.f,0.f,0.f,0.f,0.f,0.f};
                const int mrow = mt*16 + n;   // A-matrix row for this lane
#pragma unroll
                for (int kt = 0; kt < 5; ++kt) {
                    const int kb = 4*kt + 2*half;   // this lane's first K index
                    v2f Av, Bv;
                    Av.x = (mrow < F_M) ? W[mrow*F_M + kb]     : 0.f;
                    Av.y = (mrow < F_M) ? W[mrow*F_M + kb + 1] : 0.f;
                    Bv.x = (n == 0) ? v[kb]     : 0.f;
                    Bv.y = (n == 0) ? v[kb + 1] : 0.f;
                    acc = __builtin_amdgcn_wmma_f32_16x16x4_f32(
                        false, Av, false, Bv, (short)0, acc, false, false);
                }
                if (n == 0) {                 // column N=0 lives in lanes 0 and 16
                    const int base = mt*16 + half*8;
#pragma unroll
                    for (int r = 0; r < 8; ++r) cvec[mat][base + r] = acc[r];
                }
            }
        }
    }
#else
    if (lane == 0) {
        const float* Ws[4] = {Wb1_d1, Wb1_d2, Wb2_d1, Wb2_d2};
        const float* vs[4] = {Wb1_lin, Wb1_lin, Wb2_lin, Wb2_lin};
        for (int mat = 0; mat < 4; ++mat)
            for (int f = 0; f < F_M; ++f) {
                float s = 0.f;
                for (int g = 0; g < F_M; ++g) s += Ws[mat][f*F_M+g] * vs[mat][g];
                cvec[mat][f] = s;
            }
    }
#endif
    __syncthreads();
    if (lane != 0 || blockIdx.x != 0) return;

    // ---- serial remainder (tiny) ----
    // mm[f][k] = (R@m)[k][f]  -> keep as Rm[k][f]
    float Rm[3][F_M];
    for (int k = 0; k < 3; ++k)
        for (int f = 0; f < F_M; ++f) {
            float s = 0.f;
            for (int j = 0; j < 3; ++j) s += R[k*3+j] * m[j*F_M+f];
            Rm[k][f] = s;
        }

    // map_m(Wl, Wd) -> M = out^T out (sum over channel f)
    float M[4][3][3];
    const float* Wls[4] = {Wl1, Wl2, Wl3, Wl4};
    const float* Wds[4] = {Wd1, Wd2, Wd3, Wd4};
    for (int im = 0; im < 4; ++im) {
        const float* Wl = Wls[im]; const float* Wd = Wds[im];
        float y[3][3], d[3][3], o[3][3];
        for (int oo = 0; oo < 3; ++oo)
            for (int k = 0; k < 3; ++k) {
                float s = 0.f;
                for (int f = 0; f < F_M; ++f) s += Wl[oo*F_M+f] * Rm[k][f];
                y[oo][k] = s;
            }
        for (int oo = 0; oo < 3; ++oo)
            for (int k = 0; k < 3; ++k)
                d[oo][k] = Wd[oo*3+0]*y[0][k] + Wd[oo*3+1]*y[1][k] + Wd[oo*3+2]*y[2][k];
        for (int f = 0; f < 3; ++f) {
            float kxd = -2.f * (y[f][0]*d[f][0] + y[f][1]*d[f][1] + y[f][2]*d[f][2]);
            float kdd = -2.f * (d[f][0]*d[f][0] + d[f][1]*d[f][1] + d[f][2]*d[f][2]);
            float denom = fminf(kdd, -1e-12f);
            float scale = kxd / denom;
            for (int k = 0; k < 3; ++k)
                o[f][k] = (kxd < 0.f) ? y[f][k] : (y[f][k] - scale * d[f][k]);
        }
        for (int p = 0; p < 3; ++p)
            for (int q = 0; q < 3; ++q)
                M[im][p][q] = o[0][p]*o[0][q] + o[1][p]*o[1][q] + o[2][p]*o[2][q];
    }

    // u = R @ (10*[sin t, cos t, sin t])
    float tv = t[0];
    float sv = 10.f * sinf(tv), cv = 10.f * cosf(tv);
    float uu[3] = {sv, cv, sv};
    float u[3];
    for (int k = 0; k < 3; ++k)
        u[k] = R[k*3+0]*uu[0] + R[k*3+1]*uu[1] + R[k*3+2]*uu[2];

    // rank-1 channel collapses (c-vectors from the WMMA stage)
    float alpha = 0.f, beta = 0.f, alpha_u = 0.f, beta_u = 0.f;
    for (int f = 0; f < F_M; ++f) {
        alpha   += Wout[f] * cvec[0][f] * cvec[1][f];
        beta    += Wout[f] * Wb1_lin[f];
        alpha_u += Wout[f] * cvec[2][f] * cvec[3][f];
        beta_u  += Wout[f] * Wb2_lin[f];
    }

    // constant u contribution
    float v3[3], v4[3];
    for (int p = 0; p < 3; ++p) {
        v3[p] = M[2][p][0]*u[0] + M[2][p][1]*u[1] + M[2][p][2]*u[2];
        v4[p] = M[3][p][0]*u[0] + M[3][p][1]*u[1] + M[3][p][2]*u[2];
    }
    float cu[3];
    cu[0] = alpha_u*(v3[1]*v4[2] - v3[2]*v4[1]) + beta_u*u[0];
    cu[1] = alpha_u*(v3[2]*v4[0] - v3[0]*v4[2]) + beta_u*u[1];
    cu[2] = alpha_u*(v3[0]*v4[1] - v3[1]*v4[0]) + beta_u*u[2];

    for (int p = 0; p < 3; ++p)
        for (int q = 0; q < 3; ++q) {
            cst[p*3+q]     = alpha * M[0][p][q];   // A = alpha*M1
            cst[9 + p*3+q] = M[1][p][q];           // C = M2
        }
    cst[18] = beta;
    cst[19] = cu[0]; cst[20] = cu[1]; cst[21] = cu[2];
    cst[22] = 0.f;   cst[23] = 0.f;
}

// ---------------------------------------------------------------------------
// Kernel 2: streaming map over 1M states. 4 elements (= 3 float4) per thread.
// Async path: each wave issues 3 fully-contiguous global_load_async_to_lds_b128
// rounds (512B each), waits asynccnt, then lanes read their 48B slice from LDS.
// ---------------------------------------------------------------------------
__global__ __launch_bounds__(256) void ln_ode_stream(
    const float* __restrict__ x, float* __restrict__ out,
    const float* __restrict__ cst, long long nb)
{
#if HAVE_ASYNC_LDS
    __shared__ float4 sbuf[768];  // 8 waves * 96 float4 = 12 KB
#endif
    const long long ng4 = nb >> 2;        // complete 4-element groups
    const long long nf4 = ng4 * 3;        // float4 count covered by groups
    const int tid = threadIdx.x;
    const long long gid = (long long)blockIdx.x * blockDim.x + tid;

    float A[9], C[9];
#pragma unroll
    for (int i = 0; i < 9; ++i) { A[i] = cst[i]; C[i] = cst[9 + i]; }
    const float beta = cst[18];
    const float cu0 = cst[19], cu1 = cst[20], cu2 = cst[21];

    const float4* __restrict__ xf4 = (const float4*)x;
    float4* __restrict__ of4 = (float4*)out;

    float4 a, b, c;
    const bool active = (gid < ng4);

#if HAVE_ASYNC_LDS
    const int wave = tid >> 5;
    const int lane = tid & 31;
    const long long wbase = 3 * ((long long)blockIdx.x * blockDim.x + ((long long)wave << 5));
    float4* wl = &sbuf[wave * 96];
#pragma unroll
    for (int i = 0; i < 3; ++i) {
        long long f4 = wbase + i*32 + lane;
        if (f4 < nf4) {
            __builtin_amdgcn_global_load_async_to_lds_b128(
                (gptr_v4i)const_cast<float4*>(xf4 + f4),
                (lptr_v4i)(wl + i*32 + lane),
                0, 0);
        }
    }
#if __has_builtin(__builtin_amdgcn_s_wait_asynccnt)
    __builtin_amdgcn_s_wait_asynccnt(0);
#else
    asm volatile("s_wait_asynccnt 0" ::: "memory");
#endif
    asm volatile("" ::: "memory");
    if (active) {
        a = wl[3*lane + 0];
        b = wl[3*lane + 1];
        c = wl[3*lane + 2];
    }
#else
    if (active) {
        a = xf4[3*gid + 0];
        b = xf4[3*gid + 1];
        c = xf4[3*gid + 2];
    }
#endif

    if (active) {
        float xs[4][3] = {{a.x,a.y,a.z},{a.w,b.x,b.y},{b.z,b.w,c.x},{c.y,c.z,c.w}};
        float rs[4][3];
#pragma unroll
        for (int e = 0; e < 4; ++e) {
            float x0 = xs[e][0], x1 = xs[e][1], x2 = xs[e][2];
            float y10 = A[0]*x0 + A[1]*x1 + A[2]*x2;
            float y11 = A[3]*x0 + A[4]*x1 + A[5]*x2;
            float y12 = A[6]*x0 + A[7]*x1 + A[8]*x2;
            float y20 = C[0]*x0 + C[1]*x1 + C[2]*x2;
            float y21 = C[3]*x0 + C[4]*x1 + C[5]*x2;
            float y22 = C[6]*x0 + C[7]*x1 + C[8]*x2;
            rs[e][0] = y11*y22 - y12*y21 + beta*x0 + cu0;
            rs[e][1] = y12*y20 - y10*y22 + beta*x1 + cu1;
            rs[e][2] = y10*y21 - y11*y20 + beta*x2 + cu2;
        }
        of4[3*gid + 0] = make_float4(rs[0][0], rs[0][1], rs[0][2], rs[1][0]);
        of4[3*gid + 1] = make_float4(rs[1][1], rs[1][2], rs[2][0], rs[2][1]);
        of4[3*gid + 2] = make_float4(rs[2][2], rs[3][0], rs[3][1], rs[3][2]);
    }

    // tail: nb % 4 leftover elements, scalar path on the first few threads
    const long long tail_start = ng4 << 2;
    const long long ntail = nb - tail_start;
    if (gid < ntail) {
        const long long e = tail_start + gid;
        float x0 = x[3*e + 0], x1 = x[3*e + 1], x2 = x[3*e + 2];
        float y10 = A[0]*x0 + A[1]*x1 + A[2]*x2;
        float y11 = A[3]*x0 + A[4]*x1 + A[5]*x2;
        float y12 = A[6]*x0 + A[7]*x1 + A[8]*x2;
        float y20 = C[0]*x0 + C[1]*x1 + C[2]*x2;
        float y21 = C[3]*x0 + C[4]*x1 + C[5]*x2;
        float y22 = C[6]*x0 + C[7]*x1 + C[8]*x2;
        out[3*e + 0] = y11*y22 - y12*y21 + beta*x0 + cu0;
        out[3*e + 1] = y12*y20 - y10*y22 + beta*x1 + cu1;
        out[3*e + 2] = y10*y21 - y11*y20 + beta*x2 + cu2;
    }
}

// ---------------------------------------------------------------------------
extern "C" void kernel_launch(void* const* d_in, const int* in_sizes, int n_in,
                              void* d_out, int out_size, void* d_ws, size_t ws_size,
                              hipStream_t stream)
{
    const float* t       = (const float*)d_in[0];
    const float* x       = (const float*)d_in[1];
    const float* R       = (const float*)d_in[2];
    const float* m       = (const float*)d_in[3];
    const float* Wl1     = (const float*)d_in[4];
    const float* Wd1     = (const float*)d_in[5];
    const float* Wl2     = (const float*)d_in[6];
    const float* Wd2     = (const float*)d_in[7];
    const float* Wl3     = (const float*)d_in[8];
    const float* Wd3     = (const float*)d_in[9];
    const float* Wl4     = (const float*)d_in[10];
    const float* Wd4     = (const float*)d_in[11];
    const float* Wb1_lin = (const float*)d_in[12];
    const float* Wb1_d1  = (const float*)d_in[13];
    const float* Wb1_d2  = (const float*)d_in[14];
    const float* Wb2_lin = (const float*)d_in[15];
    const float* Wb2_d1  = (const float*)d_in[16];
    const float* Wb2_d2  = (const float*)d_in[17];
    const float* Wout    = (const float*)d_in[18];

    float* cst = (float*)d_ws;
    long long nb = (long long)in_sizes[1] / 3;

    ln_ode_precompute<<<1, 32, 0, stream>>>(t, R, m, Wl1, Wd1, Wl2, Wd2, Wl3, Wd3,
                                            Wl4, Wd4, Wb1_lin, Wb1_d1, Wb1_d2,
                                            Wb2_lin, Wb2_d1, Wb2_d2, Wout, cst);

    long long ng4  = nb >> 2;
    long long tail = nb - (ng4 << 2);
    long long threads = (ng4 > tail) ? ng4 : tail;
    if (threads < 1) threads = 1;
    int blocks = (int)((threads + 255) / 256);

    ln_ode_stream<<<blocks, 256, 0, stream>>>(x, (float*)d_out, cst, nb);
}